// MvnMixture_44985487459143
// MI455X (gfx1250) — compile-verified
//
#include <hip/hip_runtime.h>
#include <hip/hip_bf16.h>
#include <math.h>

typedef __attribute__((ext_vector_type(2))) float v2f;
typedef __attribute__((ext_vector_type(8))) float v8f;

#define DIM     32
#define K2      64
#define CCOMP   8
#define BATCH   2048
#define KFEAT   64           // [x | x^2]
#define NCHUNK  16           // KFEAT / 4 (K=4 per f32 WMMA)

__device__ __forceinline__ float softplus_stable(float z) {
    // log(1 + e^z) = max(z,0) + log1p(e^{-|z|})
    return fmaxf(z, 0.0f) + log1pf(expf(-fabsf(z)));
}

// Single-instruction cross-lane butterflies via ds_swizzle (group-of-32 XOR mode).
// offset = (xor_mask<<10) | (or_mask<<5) | and_mask ; and=0x1f, or=0.
__device__ __forceinline__ float swz_x1(float v) {
    return __int_as_float(__builtin_amdgcn_ds_swizzle(__float_as_int(v), 0x041F));
}
__device__ __forceinline__ float swz_x2(float v) {
    return __int_as_float(__builtin_amdgcn_ds_swizzle(__float_as_int(v), 0x081F));
}
__device__ __forceinline__ float swz_x4(float v) {
    return __int_as_float(__builtin_amdgcn_ds_swizzle(__float_as_int(v), 0x101F));
}

// grid.x = k1(4) * n_tiles(32); grid.y = 32 m-groups; block = 128 (4 waves)
// Each wave computes one 16(batch) x 16(component) tile via 16x f32 WMMA,
// then logsumexp over groups of 8 components (one k2 each).
__global__ void __launch_bounds__(128)
mvn_mixture_wmma_kernel(const float* __restrict__ inputs,   // [4,2048,32]
                        const float* __restrict__ kern,     // [4,64,8,64]
                        const float* __restrict__ mixw,     // [4,64,8]
                        float* __restrict__ out)            // [4,2048,64]
{
    __shared__ float lds_W[KFEAT * 16];   // [row(K), comp]; -0.5 pre-folded
    __shared__ float lds_pld[8 * 16];     // partial log-det
    __shared__ float lds_pc0[8 * 16];     // partial sum(w*mu^2)
    __shared__ float lds_beta[16];

    const int k1i   = blockIdx.x >> 5;        // 0..3
    const int ntile = blockIdx.x & 31;        // 0..31
    const int n0    = ntile * 16;             // first of 16 components

    const int tid = threadIdx.x;

    // ---- Stage 1a: all 128 threads build GEMM coefficients (8 thr/comp) ----
    {
        const int comp  = tid & 15;           // 0..15 within tile
        const int chunk = tid >> 4;           // 0..7, 4 dims each
        const int n  = n0 + comp;             // global component id 0..511
        const int k2 = n >> 3;
        const int c  = n & 7;
        const float* kp = kern + ((size_t)((k1i * K2 + k2) * CCOMP + c)) * (2 * DIM);

        const float SP_INV1 = 0.5413248546129181f;   // log(expm1(1))
        float pld = 0.0f, pc0 = 0.0f;
        #pragma unroll
        for (int j = 0; j < 4; ++j) {
            const int d = chunk * 4 + j;
            float mu  = kp[d];
            float raw = kp[DIM + d];
            float sd  = softplus_stable(raw + SP_INV1) + 1e-8f;
            float pin = 1.0f / sd;
            float w   = pin * pin;
            pld += logf(sd);
            pc0 += w * mu * mu;
            // -0.5 folded in: q_acc = -0.5 * (md_sq - c0) directly out of WMMA
            lds_W[d * 16 + comp]         = w * mu;        // pairs with x  (= -0.5 * -2wmu)
            lds_W[(DIM + d) * 16 + comp] = -0.5f * w;     // pairs with x^2
        }
        lds_pld[chunk * 16 + comp] = pld;
        lds_pc0[chunk * 16 + comp] = pc0;
    }
    __syncthreads();

    // ---- Stage 1b: fold bias terms (16 threads) ----
    if (tid < 16) {
        float logdet = 0.0f, c0 = 0.0f;
        #pragma unroll
        for (int ch = 0; ch < 8; ++ch) {
            logdet += lds_pld[ch * 16 + tid];
            c0     += lds_pc0[ch * 16 + tid];
        }
        logdet *= 2.0f;

        const int n  = n0 + tid;
        const int k2 = n >> 3;
        const int c  = n & 7;
        const float* mp = mixw + (size_t)(k1i * K2 + k2) * CCOMP;
        float mx = mp[0];
        #pragma unroll
        for (int j = 1; j < CCOMP; ++j) mx = fmaxf(mx, mp[j]);
        float se = 0.0f;
        #pragma unroll
        for (int j = 0; j < CCOMP; ++j) se += expf(mp[j] - mx);
        float logmix = mp[c] - mx - logf(se);

        const float CONST_TERM = 58.81206994227444f;   // 32 * log(2*pi)
        lds_beta[tid] = -0.5f * (CONST_TERM + logdet + c0) + logmix;
    }
    __syncthreads();

    // ---- Stage 2: per-wave 16x16 WMMA tile ----
    const int lane = tid & 31;
    const int wave = tid >> 5;                 // 0..3
    const int col  = lane & 15;                // M index for A, N index for B/C/D
    const int h    = lane >> 4;                // half-wave select
    const int m_base = (blockIdx.y * 4 + wave) * 16;   // batch tile base

    // B fragments from LDS: chunk kk covers K rows kk*4..kk*4+3
    v2f bf[NCHUNK];
    #pragma unroll
    for (int kk = 0; kk < NCHUNK; ++kk) {
        int r0 = kk * 4 + h * 2;
        bf[kk].x = lds_W[(r0 + 0) * 16 + col];
        bf[kk].y = lds_W[(r0 + 1) * 16 + col];
    }
    const float beta = lds_beta[col];

    // A fragments: lane holds row m = col; features kk*4 + h*2 .. +1
    const float* xp = inputs + ((size_t)(k1i * BATCH + m_base + col)) * DIM + h * 2;
    v2f af[NCHUNK];
    #pragma unroll
    for (int kk = 0; kk < 8; ++kk)
        af[kk] = *(const v2f*)(xp + kk * 4);          // global_load_b64
    #pragma unroll
    for (int kk = 0; kk < 8; ++kk)
        af[8 + kk] = af[kk] * af[kk];                 // x^2 features

    v8f acc = {0.f, 0.f, 0.f, 0.f, 0.f, 0.f, 0.f, 0.f};
    #pragma unroll
    for (int kk = 0; kk < NCHUNK; ++kk) {
        acc = __builtin_amdgcn_wmma_f32_16x16x4_f32(
            /*neg_a=*/false, af[kk],
            /*neg_b=*/false, bf[kk],
            /*c_mod=*/(short)0, acc,
            /*reuse_a=*/false, /*reuse_b=*/false);
    }

    // ---- Epilogue: s = acc + beta (=-0.5*md_sq + bias); logsumexp across
    //      8-lane component groups via single-instruction ds_swizzle butterflies.
    const int k2_out = (n0 >> 3) + ((lane >> 3) & 1);
    const bool store_lane = ((lane & 7) == 0);
    #pragma unroll
    for (int r = 0; r < 8; ++r) {
        float s = acc[r] + beta;
        float m = s;
        m = fmaxf(m, swz_x1(m));
        m = fmaxf(m, swz_x2(m));
        m = fmaxf(m, swz_x4(m));
        float e = expf(s - m);
        e += swz_x1(e);
        e += swz_x2(e);
        e += swz_x4(e);
        float res = m + logf(e);
        if (store_lane) {
            int b = m_base + h * 8 + r;
            out[((size_t)(k1i * BATCH + b)) * K2 + k2_out] = res;
        }
    }
}

extern "C" void kernel_launch(void* const* d_in, const int* in_sizes, int n_in,
                              void* d_out, int out_size, void* d_ws, size_t ws_size,
                              hipStream_t stream) {
    (void)in_sizes; (void)n_in; (void)out_size; (void)d_ws; (void)ws_size;
    const float* inputs = (const float*)d_in[0];   // [4,2048,32]
    const float* kern   = (const float*)d_in[1];   // [4,64,8,64]
    const float* mixw   = (const float*)d_in[2];   // [4,64,8]
    float* out          = (float*)d_out;           // [4,2048,64]

    dim3 grid(4 * 32, 32, 1);   // (k1 x n_tiles, m_groups)
    dim3 block(128, 1, 1);      // 4 waves, one 16x16 tile each
    mvn_mixture_wmma_kernel<<<grid, block, 0, stream>>>(inputs, kern, mixw, out);
}